// constrained_ouroboros_69492570849416
// MI455X (gfx1250) — compile-verified
//
#include <hip/hip_runtime.h>
#include <math.h>

// ---- problem constants (from reference) ----
#define BATCH 4
#define SEQ   8192
#define DCH   64          // D_CONTROL == D_INNER
#define DSTATE 16
#define DCONV 4
#define DTR   4
#define POLY  40
#define NPB   (2*POLY*POLY)     // 3200
#define KST   63                // states channels
#define TAUC  1000.0f
#define NTOK  (BATCH*SEQ)       // 32768
#define YHATN (BATCH*(SEQ-1)*2) // 65528

typedef __attribute__((ext_vector_type(2))) float v2f;
typedef __attribute__((ext_vector_type(8))) float v8f;

__device__ __forceinline__ float siluf(float v) { return v / (1.0f + expf(-v)); }
__device__ __forceinline__ float softplusf(float v) { return (v > 20.0f) ? v : log1pf(expf(v)); }

// ---------------- kernel 0: zero-pad b_net_w rows 63 -> 64 (one-time, 0.8 MB, L2-resident) ----
__global__ __launch_bounds__(256) void pad_bnw(
    const float* __restrict__ bnw, float* __restrict__ wpad)
{
  long idx = (long)blockIdx.x * blockDim.x + threadIdx.x;
  if (idx >= (long)NPB * 64) return;
  long m = idx >> 6;
  int  c = (int)(idx & 63);
  wpad[idx] = (c < KST) ? bnw[m*KST + c] : 0.0f;
}

// ---------------- kernel 1: build u = x_in @ control_proj_w.T + b ----------------
__global__ __launch_bounds__(256) void build_u(
    const float* __restrict__ x, const float* __restrict__ y,
    const float* __restrict__ cpw, const float* __restrict__ cpb,
    float* __restrict__ u)
{
  long idx = (long)blockIdx.x * blockDim.x + threadIdx.x;
  if (idx >= (long)NTOK * DCH) return;
  long tok = idx >> 6;
  int  c   = (int)(idx & 63);
  long b = tok / SEQ, l = tok % SEQ;
  long lf = SEQ - 1 - l;
  float xv = x[b*SEQ + l],  yvv = y[b*SEQ + l];
  float xf = x[b*SEQ + lf], yf  = y[b*SEQ + lf];
  // x_in = [xdot, y, flip(xdot), flip(y)]
  float i0 = yvv - xv, i1 = yvv, i2 = yf - xf, i3 = yf;
  u[idx] = cpw[c*4+0]*i0 + cpw[c*4+1]*i1 + cpw[c*4+2]*i2 + cpw[c*4+3]*i3 + cpb[c];
}

// ---------------- kernel 2: rmsnorm + in_proj (64 -> 128), one token per block ----------------
__global__ __launch_bounds__(128) void rms_inproj(
    const float* __restrict__ u, const float* __restrict__ normw,
    const float* __restrict__ inw, float* __restrict__ xc, float* __restrict__ zb)
{
  __shared__ float un[DCH];
  __shared__ float sq[DCH];
  __shared__ float rinv;
  long tok = blockIdx.x;
  int tid = threadIdx.x;
  float v = 0.0f;
  if (tid < DCH) { v = u[tok*DCH + tid]; sq[tid] = v*v; }
  __syncthreads();
  if (tid == 0) {
    float s = 0.0f;
    for (int i = 0; i < DCH; ++i) s += sq[i];
    rinv = rsqrtf(s * (1.0f/DCH) + 1e-5f);
  }
  __syncthreads();
  if (tid < DCH) un[tid] = v * rinv * normw[tid];
  __syncthreads();
  float acc = 0.0f;
  const float* wr = inw + (long)tid*DCH;   // in_proj_w: (128,64) row-major
  for (int c = 0; c < DCH; ++c) acc += wr[c]*un[c];
  if (tid < DCH) xc[tok*DCH + tid] = acc;
  else           zb[tok*DCH + (tid-DCH)] = acc;
}

// ---------------- kernel 3: causal depthwise conv4 + silu + x_proj + dt_proj ----------------
__global__ __launch_bounds__(64) void conv_xproj(
    const float* __restrict__ xc, const float* __restrict__ convw,
    const float* __restrict__ convb, const float* __restrict__ xpw,
    const float* __restrict__ dtw, const float* __restrict__ dtb,
    float* __restrict__ xss, float* __restrict__ delt,
    float* __restrict__ Bm, float* __restrict__ Cm)
{
  __shared__ float xs[DCH];
  __shared__ float dbc[DTR + 2*DSTATE];
  long tok = blockIdx.x;
  long b = tok / SEQ, l = tok % SEQ;
  int e = threadIdx.x;
  float a = convb[e];
  #pragma unroll
  for (int k = 0; k < DCONV; ++k) {
    long ll = l - (DCONV-1) + k;
    if (ll >= 0) a += convw[e*DCONV + k] * xc[(b*SEQ + ll)*DCH + e];
  }
  float s = siluf(a);
  xs[e] = s;
  xss[tok*DCH + e] = s;
  __syncthreads();
  if (e < DTR + 2*DSTATE) {
    float d = 0.0f;
    const float* wr = xpw + (long)e*DCH;   // x_proj_w: (36,64)
    for (int c = 0; c < DCH; ++c) d += wr[c]*xs[c];
    dbc[e] = d;
  }
  __syncthreads();
  float dv = dtb[e];
  #pragma unroll
  for (int r = 0; r < DTR; ++r) dv += dtw[e*DTR + r]*dbc[r];
  delt[tok*DCH + e] = softplusf(dv);
  if (e < DSTATE) Bm[tok*DSTATE + e] = dbc[DTR + e];
  if (e < DSTATE) Cm[tok*DSTATE + e] = dbc[DTR + DSTATE + e];
}

// ---------------- kernel 4: SSM sequential scan + D skip + silu(z) gating ----------------
// one wave per (batch, channel-pair); lane = eSub*16 + n (n = state index)
__global__ __launch_bounds__(32) void ssm_scan(
    const float* __restrict__ xss, const float* __restrict__ delt,
    const float* __restrict__ Bm, const float* __restrict__ Cm,
    const float* __restrict__ zb, const float* __restrict__ alog,
    const float* __restrict__ Dvec, float* __restrict__ yv)
{
  int lane = threadIdx.x;
  int half = lane >> 4;
  int n = lane & 15;
  int b = blockIdx.x >> 5;
  int epair = blockIdx.x & 31;
  int e = epair*2 + half;
  float Aen = -expf(alog[e*DSTATE + n]);
  float Dp = Dvec[e];
  float h = 0.0f;
  for (long l = 0; l < SEQ; ++l) {
    long base = (long)b*SEQ + l;
    float dl  = delt[base*DCH + e];
    float xsv = xss[base*DCH + e];
    float bm  = Bm[base*DSTATE + n];
    float cm  = Cm[base*DSTATE + n];
    h = expf(dl*Aen)*h + dl*bm*xsv;
    float p = h*cm;                 // reduce over 16 states (stays within half-wave)
    p += __shfl_xor(p, 1, 32);
    p += __shfl_xor(p, 2, 32);
    p += __shfl_xor(p, 4, 32);
    p += __shfl_xor(p, 8, 32);
    if (n == 0) {
      float yss = p + xsv*Dp;
      float zv = zb[base*DCH + e];
      yv[base*DCH + e] = yss * siluf(zv);
    }
  }
}

// ---------------- kernel 5: out_proj (64->64) + residual ----------------
__global__ __launch_bounds__(64) void outproj_res(
    const float* __restrict__ yv, const float* __restrict__ outw,
    float* __restrict__ u)
{
  __shared__ float yl[DCH];
  long tok = blockIdx.x;
  int c = threadIdx.x;
  yl[c] = yv[tok*DCH + c];
  __syncthreads();
  float acc = 0.0f;
  const float* wr = outw + (long)c*DCH;
  for (int e = 0; e < DCH; ++e) acc += wr[e]*yl[e];
  u[tok*DCH + c] += acc;
}

// ---------------- kernel 6: states = flip(u)[:, :, 1:] ----------------
__global__ __launch_bounds__(256) void write_states(
    const float* __restrict__ u, float* __restrict__ states)
{
  long idx = (long)blockIdx.x*blockDim.x + threadIdx.x;
  if (idx >= (long)NTOK*KST) return;
  long tok = idx / KST;
  int  c   = (int)(idx % KST);
  long b = tok / SEQ, l = tok % SEQ;
  states[idx] = u[(b*SEQ + (SEQ-1-l))*DCH + (c+1)];
}

// ---------------- kernel 7: fused b-GEMM (WMMA f32 16x16x4) + poly contraction ----------------
// One wave per 16-token tile; streams all 200 N-tiles of the padded weight (819 KB,
// L2-resident), contracting each 16x16 b-tile with q = z0^j * z1^k in registers.
// Overridden entries (b[0,1,0], b[1,0,1], b[1,1,0]) handled as rank-1 corrections.
__global__ __launch_bounds__(32) void fused_bnet_poly(
    const float* __restrict__ states, const float* __restrict__ xin,
    const float* __restrict__ yin, const float* __restrict__ wpad,
    const float* __restrict__ bnb, const float* __restrict__ omw,
    const float* __restrict__ omb, float* __restrict__ yhat)
{
  __shared__ float S[16][DCH];      // 16 tokens x 64 (K padded 63->64 with zero)
  __shared__ float P1[16][POLY];
  __shared__ float P2[16][POLY];
  __shared__ float omg[16];
  __shared__ float bspec[16][3];    // computed b at the three overridden indices

  const int lane = threadIdx.x;
  const int half = lane >> 4;
  const int r = lane & 15;
  const long t0 = (long)blockIdx.x * 16;

  #pragma unroll
  for (int i = 0; i < 32; ++i) {          // cooperative, coalesced S-tile load
    int flat = i*32 + lane;
    int row = flat >> 6, col = flat & 63;
    long t = t0 + row;
    S[row][col] = (col < KST) ? states[t*KST + col] : 0.0f;
  }
  __syncthreads();

  if (lane < 16) {
    long t = t0 + lane;
    long b = t / SEQ, l = t % SEQ;
    float z0 = xin[b*SEQ + l];            // z = [x, y-x]
    float z1 = yin[b*SEQ + l] - z0;
    float p1 = 1.0f, p2 = 1.0f;
    for (int j = 0; j < POLY; ++j) { P1[lane][j] = p1; P2[lane][j] = p2; p1 *= z0; p2 *= z1; }
    float om = omb[0];
    for (int c = 0; c < KST; ++c) om += S[lane][c]*omw[c];
    omg[lane] = om * TAUC;
    // computed b at the three overridden (d,j,k) positions, for this token
    const int mspec[3] = {40, 1601, 1640};   // (0,1,0), (1,0,1), (1,1,0)
    #pragma unroll
    for (int s = 0; s < 3; ++s) {
      const float* wr = wpad + (long)mspec[s]*64;
      float acc = bnb[mspec[s]];
      for (int c = 0; c < KST; ++c) acc += S[lane][c]*wr[c];
      bspec[lane][s] = TAUC * acc;
    }
  }
  __syncthreads();

  // A fragments for V_WMMA_F32_16X16X4_F32 (ISA 7.12.2: lane r holds M=r, VGPR i -> K=2*half+i)
  float Af[32];
  #pragma unroll
  for (int c = 0; c < 16; ++c) {
    Af[2*c]   = S[r][4*c + 2*half];
    Af[2*c+1] = S[r][4*c + 2*half + 1];
  }

  float yacc0[8], yacc1[8];
  #pragma unroll
  for (int v = 0; v < 8; ++v) { yacc0[v] = 0.0f; yacc1[v] = 0.0f; }

  for (int nt = 0; nt < NPB/16; ++nt) {
    int m = nt*16 + r;                       // b_net output column, fixed per lane
    const float* wrow = wpad + (long)m*64;   // padded weight row (64 floats, col 63 = 0)
    if (nt + 1 < NPB/16) __builtin_prefetch(wrow + 16*64, 0, 1);
    v8f acc = {0.f,0.f,0.f,0.f,0.f,0.f,0.f,0.f};
    #pragma unroll
    for (int c = 0; c < 16; ++c) {           // K = 64 in chunks of 4
      int kb = 4*c + 2*half;
      v2f a;  a.x = Af[2*c];  a.y = Af[2*c+1];
      v2f bb = *(const v2f*)(wrow + kb);     // unconditional 8B load -> global_load_b64
      acc = __builtin_amdgcn_wmma_f32_16x16x4_f32(
              false, a, false, bb, (short)0, acc, false, false);
    }
    float biasm = bnb[m];
    int d = m / 1600;
    int rem = m - d*1600;
    int j = rem / POLY;
    int k = rem - j*POLY;
    #pragma unroll
    for (int v = 0; v < 8; ++v) {            // C layout: lanes0-15 row v, lanes16-31 row v+8
      int tok = v + 8*half;
      float bv = TAUC * (acc[v] + biasm);
      float q = P1[tok][j] * P2[tok][k];
      if (d == 0) yacc0[v] += bv*q; else yacc1[v] += bv*q;
    }
  }
  #pragma unroll
  for (int v = 0; v < 8; ++v) {              // reduce over 16 N-lanes per half
    for (int mk = 1; mk < 16; mk <<= 1) {
      yacc0[v] += __shfl_xor(yacc0[v], mk, 32);
      yacc1[v] += __shfl_xor(yacc1[v], mk, 32);
    }
  }
  if (r == 0) {
    for (int v = 0; v < 8; ++v) {
      int tok = v + 8*half;
      long t = t0 + tok;
      long b = t / SEQ, l = t % SEQ;
      if (l == 0) continue;                  // yhat[:, 1:, :]
      float z0 = P1[tok][1];
      float z1 = P2[tok][1];
      float om = omg[tok];
      float y0 = yacc0[v] - bspec[tok][0]*z0;               // b[0,1,0] := 0
      float y1 = yacc1[v] + (1.0f - bspec[tok][1])*z1;      // b[1,0,1] := 1
      y1 += (-om*om - bspec[tok][2])*z0;                    // b[1,1,0] := -omega^2
      yhat[(b*(SEQ-1) + (l-1))*2 + 0] = y0;
      yhat[(b*(SEQ-1) + (l-1))*2 + 1] = y1;
    }
  }
}

extern "C" void kernel_launch(void* const* d_in, const int* in_sizes, int n_in,
                              void* d_out, int out_size, void* d_ws, size_t ws_size,
                              hipStream_t stream)
{
  const float *x, *y, *cpw, *cpb, *bnw, *bnb, *omw, *omb;
  const float *normw[2], *inw[2], *convw[2], *convb[2], *xpw[2],
              *dtw[2], *dtb[2], *alog[2], *Dv[2], *outw[2];
  auto F = [&](int i){ return (const float*)d_in[i]; };

  if (n_in >= 28 && in_sizes[0] == NTOK && in_sizes[1] == NTOK) {
    // setup_inputs() insertion order
    x = F(0); y = F(1);
    for (int lyr = 0; lyr < 2; ++lyr) {
      int o = 2 + lyr*10;
      normw[lyr]=F(o+0); inw[lyr]=F(o+1); convw[lyr]=F(o+2); convb[lyr]=F(o+3);
      xpw[lyr]=F(o+4); dtw[lyr]=F(o+5); dtb[lyr]=F(o+6); alog[lyr]=F(o+7);
      Dv[lyr]=F(o+8); outw[lyr]=F(o+9);
    }
    cpw=F(22); cpb=F(23); bnw=F(24); bnb=F(25); omw=F(26); omb=F(27);
  } else {
    // jax pytree sorted-key order
    bnb=F(0); bnw=F(1); cpb=F(2); cpw=F(3);
    for (int lyr = 0; lyr < 2; ++lyr) {
      int o = 4 + lyr*10;
      alog[lyr]=F(o+0); Dv[lyr]=F(o+1); convb[lyr]=F(o+2); convw[lyr]=F(o+3);
      dtb[lyr]=F(o+4); dtw[lyr]=F(o+5); inw[lyr]=F(o+6); normw[lyr]=F(o+7);
      outw[lyr]=F(o+8); xpw[lyr]=F(o+9);
    }
    omb=F(24); omw=F(25); x=F(26); y=F(27);
  }

  // workspace layout (~56 MB of fp32)
  float* w    = (float*)d_ws;
  float* u    = w;
  float* xc   = w + (size_t)NTOK*DCH;
  float* zb   = w + (size_t)NTOK*DCH*2;
  float* xss  = w + (size_t)NTOK*DCH*3;
  float* delt = w + (size_t)NTOK*DCH*4;
  float* yv   = w + (size_t)NTOK*DCH*5;
  float* Bmb  = w + (size_t)NTOK*DCH*6;
  float* Cmb  = Bmb + (size_t)NTOK*DSTATE;
  float* wpad = Cmb + (size_t)NTOK*DSTATE;   // 3200*64 floats, zero-padded b_net_w

  float* yhat   = (float*)d_out;
  float* states = yhat + YHATN;

  pad_bnw<<<(NPB*64 + 255)/256, 256, 0, stream>>>(bnw, wpad);
  build_u<<<(NTOK*DCH + 255)/256, 256, 0, stream>>>(x, y, cpw, cpb, u);
  for (int lyr = 0; lyr < 2; ++lyr) {
    rms_inproj<<<NTOK, 128, 0, stream>>>(u, normw[lyr], inw[lyr], xc, zb);
    conv_xproj<<<NTOK, 64, 0, stream>>>(xc, convw[lyr], convb[lyr], xpw[lyr],
                                        dtw[lyr], dtb[lyr], xss, delt, Bmb, Cmb);
    ssm_scan<<<BATCH*32, 32, 0, stream>>>(xss, delt, Bmb, Cmb, zb, alog[lyr], Dv[lyr], yv);
    outproj_res<<<NTOK, 64, 0, stream>>>(yv, outw[lyr], u);
  }
  write_states<<<(int)(((long)NTOK*KST + 255)/256), 256, 0, stream>>>(u, states);
  fused_bnet_poly<<<NTOK/16, 32, 0, stream>>>(states, x, y, wpad, bnb, omw, omb, yhat);
}